// SelfMultiHeadAttention_30296699306110
// MI455X (gfx1250) — compile-verified
//
#include <hip/hip_runtime.h>
#include <hip/hip_bf16.h>
#include <math.h>
#include <stdint.h>

// Problem constants (B,T,D,H from the reference)
#define BATCH   32
#define T_SEQ   1024
#define DMODEL  128
#define NHEAD   4
#define DHEAD   32
#define QKV_LD  (3 * DMODEL)          // 384
#define SCALE   0.17677669529663687f  // 1/sqrt(32)
#define NEGV    (-4294967295.0f)      // -(2^32)+1, reference padding value
#define LN_EPS  1e-9f

#define STRIDE32  36   // LDS row stride (floats) for 16x32 tiles: 144B, 16B-aligned, conflict-free
#define STRIDE128 132  // LDS row stride (floats) for 16x128 tiles: 528B, 16B-aligned, conflict-free

typedef __attribute__((ext_vector_type(2))) float v2f;
typedef __attribute__((ext_vector_type(8))) float v8f;
typedef int v4i __attribute__((vector_size(4 * sizeof(int))));
typedef __attribute__((address_space(1))) v4i* gv4i_p;  // global int4*
typedef __attribute__((address_space(3))) v4i* lv4i_p;  // LDS int4*

// ---------------------------------------------------------------------------
// f32 WMMA: D(16x16) = A(16x4) * B(4x16) + C   (V_WMMA_F32_16X16X4_F32)
// ---------------------------------------------------------------------------
static __device__ __forceinline__ v8f wmma4(v2f a, v2f b, v8f c) {
  return __builtin_amdgcn_wmma_f32_16x16x4_f32(
      /*neg_a=*/false, a, /*neg_b=*/false, b,
      /*c_mod=*/(short)0, c, /*reuse_a=*/false, /*reuse_b=*/false);
}

// ---------------------------------------------------------------------------
// Async global->LDS copy (16B per lane), tracked by ASYNCcnt
// ---------------------------------------------------------------------------
static __device__ __forceinline__ uint32_t lds_off(const void* p) {
  // generic pointers into LDS carry the byte offset in their low 32 bits
  return (uint32_t)(uintptr_t)p;
}

static __device__ __forceinline__ void async_copy16(const float* __restrict__ g,
                                                    uint32_t lds_byte_off) {
#if __has_builtin(__builtin_amdgcn_global_load_async_to_lds_b128)
  __builtin_amdgcn_global_load_async_to_lds_b128(
      (gv4i_p)(uintptr_t)g,
      (lv4i_p)(uintptr_t)lds_byte_off,
      /*imm offset=*/0, /*cpol=*/0);
#else
  asm volatile("global_load_async_to_lds_b128 %0, %1, off"
               :: "v"(lds_byte_off), "v"((uint64_t)(uintptr_t)g) : "memory");
#endif
}

static __device__ __forceinline__ void wait_async0() {
#if __has_builtin(__builtin_amdgcn_s_wait_asynccnt)
  __builtin_amdgcn_s_wait_asynccnt(0);
#else
  asm volatile("s_wait_asynccnt 0x0" ::: "memory");
#endif
  asm volatile("" ::: "memory");
}

// Stage a 16-row x 32-float tile (rows contiguous 128B in global, ld = row stride)
// into LDS with stride STRIDE32. 4 async instructions, 128B bursts per 8 lanes.
static __device__ __forceinline__ void stage_16x32(const float* __restrict__ g, int ld,
                                                   uint32_t base, int lane) {
#pragma unroll
  for (int i = 0; i < 4; ++i) {
    const int seg = i * 32 + lane;        // 128 segments of 16B
    const int row = seg >> 3, s = seg & 7;
    async_copy16(g + (size_t)row * ld + s * 4,
                 base + (uint32_t)(row * (STRIDE32 * 4) + s * 16));
  }
}

// Stage a 16-row x 128-float tile (rows contiguous 512B in global) into LDS with
// stride STRIDE128. 16 async instructions, each one fully-coalesced 512B row.
static __device__ __forceinline__ void stage_16x128(const float* __restrict__ g, int ld,
                                                    uint32_t base, int lane) {
#pragma unroll
  for (int i = 0; i < 16; ++i)
    async_copy16(g + (size_t)i * ld + lane * 4,
                 base + (uint32_t)(i * (STRIDE128 * 4) + lane * 16));
}

// ---------------------------------------------------------------------------
// Fragment gathers
// ---------------------------------------------------------------------------
// A-layout fragment (also B-layout for a transposed row-major operand, e.g. K^T):
//   lanes 0-15 : row base+lane, cols col+0,1 ; lanes 16-31: cols col+2,3
static __device__ __forceinline__ v2f frag_T_lds(const float* base, int stride,
                                                 int col, int lane) {
  const float* p = base + (lane & 15) * stride + col + ((lane >> 4) << 1);
  v2f r; r.x = p[0]; r.y = p[1];
  return r;
}
// B-layout fragment from a row-major operand used directly:
//   VGPR0 rows k0 / k0+2 , VGPR1 rows k0+1 / k0+3 , lane&15 = column
static __device__ __forceinline__ v2f fragB_lds(const float* base, int stride,
                                                int krow, int col, int lane) {
  const float* p = base + (krow + ((lane >> 4) << 1)) * stride + col + (lane & 15);
  v2f r; r.x = p[0]; r.y = p[stride];
  return r;
}
// B-layout fragment straight from global (for L2-resident weights)
static __device__ __forceinline__ v2f fragB_g(const float* __restrict__ p, int ld,
                                              int krow, int col, int lane) {
  const float* q = p + (size_t)(krow + ((lane >> 4) << 1)) * ld + col + (lane & 15);
  v2f r; r.x = q[0]; r.y = q[ld];
  return r;
}

// Row reductions across the 16-lane half holding one accumulator row
static __device__ __forceinline__ float hred_max(float v) {
  v = fmaxf(v, __shfl_xor(v, 1, 32));
  v = fmaxf(v, __shfl_xor(v, 2, 32));
  v = fmaxf(v, __shfl_xor(v, 4, 32));
  v = fmaxf(v, __shfl_xor(v, 8, 32));
  return v;
}
static __device__ __forceinline__ float hred_sum(float v) {
  v += __shfl_xor(v, 1, 32);
  v += __shfl_xor(v, 2, 32);
  v += __shfl_xor(v, 4, 32);
  v += __shfl_xor(v, 8, 32);
  return v;
}

// ---------------------------------------------------------------------------
// Kernel 1: QKV projection  qkv[32768,384] = x[32768,128] @ W[128,384]
// X tile staged async->LDS (coalesced); W fragments from L2.
// ---------------------------------------------------------------------------
__global__ __launch_bounds__(32) void qkv_gemm_kernel(const float* __restrict__ X,
                                                      const float* __restrict__ W,
                                                      float* __restrict__ QKV) {
  __shared__ float sX[16][STRIDE128];
  const int row0 = blockIdx.x * 16;
  const int col0 = blockIdx.y * 64;
  const int lane = threadIdx.x & 31;

  stage_16x128(X + (size_t)row0 * DMODEL, DMODEL, lds_off(&sX[0][0]), lane);
  wait_async0();
  __syncthreads();

  v8f acc[4] = {};
#pragma unroll 4
  for (int k = 0; k < DMODEL; k += 4) {
    v2f a = frag_T_lds(&sX[0][0], STRIDE128, k, lane);
#pragma unroll
    for (int t = 0; t < 4; ++t) {
      v2f b = fragB_g(W, QKV_LD, k, col0 + 16 * t, lane);
      acc[t] = wmma4(a, b, acc[t]);
    }
  }
  const int n = lane & 15, hs = lane >> 4;
#pragma unroll
  for (int t = 0; t < 4; ++t)
#pragma unroll
    for (int r = 0; r < 8; ++r)
      QKV[(size_t)(row0 + r + 8 * hs) * QKV_LD + col0 + 16 * t + n] = acc[t][r];
}

// ---------------------------------------------------------------------------
// Kernel 2: flash attention. One wave per (16-query tile, batch, head).
// K/V 16-key chunks double-buffered via async->LDS; S = Q*K^T (8 wmma),
// online softmax with causal + keys_length masking, P via LDS, O += P*V (8 wmma).
// ---------------------------------------------------------------------------
__global__ __launch_bounds__(32) void attn_kernel(const float* __restrict__ QKV,
                                                  const int* __restrict__ keys_length,
                                                  float* __restrict__ AOUT) {
  __shared__ float sQ[16][STRIDE32];
  __shared__ float sK[2][16][STRIDE32];
  __shared__ float sV[2][16][STRIDE32];
  __shared__ float sP[16][20];

  const int q0 = blockIdx.x * 16;
  const int b  = blockIdx.y;
  const int h  = blockIdx.z;
  const int lane = threadIdx.x & 31;
  const int n = lane & 15, hs = lane >> 4;

  const float* Q = QKV + (size_t)b * T_SEQ * QKV_LD + h * DHEAD;
  const float* K = QKV + (size_t)b * T_SEQ * QKV_LD + DMODEL + h * DHEAD;
  const float* V = QKV + (size_t)b * T_SEQ * QKV_LD + 2 * DMODEL + h * DHEAD;

  const int kl = keys_length[b];              // uniform per wave
  const int maxkey = min(q0 + 15, kl - 1);    // causal + padding bound
  const int nchunk = (maxkey >> 4) + 1;       // >= 1 always (kl >= 1)

  // stage Q tile and first K/V chunk
  stage_16x32(Q + (size_t)q0 * QKV_LD, QKV_LD, lds_off(&sQ[0][0]), lane);
  stage_16x32(K, QKV_LD, lds_off(&sK[0][0][0]), lane);
  stage_16x32(V, QKV_LD, lds_off(&sV[0][0][0]), lane);
  wait_async0();
  __syncthreads();

  v2f qf[8];
#pragma unroll
  for (int c = 0; c < 8; ++c) qf[c] = frag_T_lds(&sQ[0][0], STRIDE32, 4 * c, lane);

  float m_i[8], l_i[8];
  v8f o0 = {}, o1 = {};
#pragma unroll
  for (int r = 0; r < 8; ++r) { m_i[r] = -INFINITY; l_i[r] = 0.f; }

  int buf = 0;
  for (int kc = 0; kc < nchunk; ++kc) {
    const int key0 = kc * 16;
    const bool more = (kc + 1 < nchunk);
    if (more) {  // prefetch next chunk into the other buffer while we compute
      stage_16x32(K + (size_t)(key0 + 16) * QKV_LD, QKV_LD, lds_off(&sK[buf ^ 1][0][0]), lane);
      stage_16x32(V + (size_t)(key0 + 16) * QKV_LD, QKV_LD, lds_off(&sV[buf ^ 1][0][0]), lane);
    }

    // S = Q * K^T over dh=32 (8 K-chunks of 4)
    v8f s = {};
#pragma unroll
    for (int c = 0; c < 8; ++c) {
      v2f kf = frag_T_lds(&sK[buf][0][0], STRIDE32, 4 * c, lane);  // K^T as B
      s = wmma4(qf[c], kf, s);
    }

    // scale, mask, online softmax update (row r lives in the 16-lane half)
    const int key = key0 + n;
    const bool kvalid = (key < kl);
    float corr[8];
#pragma unroll
    for (int r = 0; r < 8; ++r) {
      const int qrow = q0 + r + 8 * hs;
      float sv = s[r] * SCALE;
      sv = (kvalid && key <= qrow) ? sv : NEGV;
      const float mnew = fmaxf(m_i[r], hred_max(sv));
      const float pv = __expf(sv - mnew);
      corr[r] = __expf(m_i[r] - mnew);
      l_i[r] = l_i[r] * corr[r] + hred_sum(pv);
      m_i[r] = mnew;
      sP[r + 8 * hs][n] = pv;  // C-layout -> LDS for relayout to A-layout
    }
#pragma unroll
    for (int r = 0; r < 8; ++r) { o0[r] *= corr[r]; o1[r] *= corr[r]; }
    __syncthreads();

    // O += P * V  (contraction over the 16 keys: 4 chunks x 2 dh-tiles)
#pragma unroll
    for (int c = 0; c < 4; ++c) {
      v2f pf;
      const int pcol = 4 * c + 2 * hs;
      pf.x = sP[n][pcol];
      pf.y = sP[n][pcol + 1];
      v2f vb0 = fragB_lds(&sV[buf][0][0], STRIDE32, 4 * c, 0, lane);
      v2f vb1 = fragB_lds(&sV[buf][0][0], STRIDE32, 4 * c, 16, lane);
      o0 = wmma4(pf, vb0, o0);
      o1 = wmma4(pf, vb1, o1);
    }
    __syncthreads();
    if (more) { wait_async0(); buf ^= 1; }
  }

  // normalize and store head output into [B*T, 128] (heads concatenated)
  float* out = AOUT + (size_t)b * T_SEQ * DMODEL + h * DHEAD;
#pragma unroll
  for (int r = 0; r < 8; ++r) {
    const float inv = 1.f / l_i[r];
    const int qrow = q0 + r + 8 * hs;
    out[(size_t)qrow * DMODEL + n]      = o0[r] * inv;
    out[(size_t)qrow * DMODEL + 16 + n] = o1[r] * inv;
  }
}

// ---------------------------------------------------------------------------
// Kernel 3: output projection + residual + LayerNorm, fused per 16-row tile.
// attn tile staged async->LDS; Wo fragments from L2.
// ---------------------------------------------------------------------------
__global__ __launch_bounds__(32) void out_ln_kernel(const float* __restrict__ A,
                                                    const float* __restrict__ Wo,
                                                    const float* __restrict__ X,
                                                    const float* __restrict__ gamma,
                                                    const float* __restrict__ beta,
                                                    float* __restrict__ Out) {
  __shared__ float sA[16][STRIDE128];
  const int row0 = blockIdx.x * 16;
  const int lane = threadIdx.x & 31;
  const int n = lane & 15, hs = lane >> 4;

  stage_16x128(A + (size_t)row0 * DMODEL, DMODEL, lds_off(&sA[0][0]), lane);
  wait_async0();
  __syncthreads();

  v8f acc[8] = {};
#pragma unroll 4
  for (int k = 0; k < DMODEL; k += 4) {
    v2f a = frag_T_lds(&sA[0][0], STRIDE128, k, lane);
#pragma unroll
    for (int t = 0; t < 8; ++t) {
      v2f b = fragB_g(Wo, DMODEL, k, t * 16, lane);
      acc[t] = wmma4(a, b, acc[t]);
    }
  }
#pragma unroll
  for (int r = 0; r < 8; ++r) {
    const int row = row0 + r + 8 * hs;
    float sum = 0.f, sumsq = 0.f;
#pragma unroll
    for (int t = 0; t < 8; ++t) {
      const float v = acc[t][r] + X[(size_t)row * DMODEL + t * 16 + n];
      acc[t][r] = v;
      sum += v;
      sumsq += v * v;
    }
    sum = hred_sum(sum);
    sumsq = hred_sum(sumsq);
    const float mu = sum * (1.f / DMODEL);
    const float var = sumsq * (1.f / DMODEL) - mu * mu;
    const float rstd = rsqrtf(var + LN_EPS);
#pragma unroll
    for (int t = 0; t < 8; ++t) {
      const int col = t * 16 + n;
      Out[(size_t)row * DMODEL + col] = (acc[t][r] - mu) * rstd * gamma[col] + beta[col];
    }
  }
}

// ---------------------------------------------------------------------------
extern "C" void kernel_launch(void* const* d_in, const int* in_sizes, int n_in,
                              void* d_out, int out_size, void* d_ws, size_t ws_size,
                              hipStream_t stream) {
  const float* x           = (const float*)d_in[0];
  const int*   keys_length = (const int*)d_in[1];
  const float* W           = (const float*)d_in[2];
  const float* Wo          = (const float*)d_in[3];
  const float* gamma       = (const float*)d_in[4];
  const float* beta        = (const float*)d_in[5];
  float* out = (float*)d_out;

  float* qkv  = (float*)d_ws;                          // 32768*384 f32 (48MB)
  float* attn = qkv + (size_t)BATCH * T_SEQ * QKV_LD;  // 32768*128 f32 (16MB)

  // 1) QKV projection: 2048 row-tiles x 6 col-groups of 64
  qkv_gemm_kernel<<<dim3((BATCH * T_SEQ) / 16, QKV_LD / 64), 32, 0, stream>>>(x, W, qkv);
  // 2) flash attention: (T/16, B, H) waves
  attn_kernel<<<dim3(T_SEQ / 16, BATCH, NHEAD), 32, 0, stream>>>(qkv, keys_length, attn);
  // 3) output projection + residual + LayerNorm
  out_ln_kernel<<<dim3((BATCH * T_SEQ) / 16), 32, 0, stream>>>(attn, Wo, x, gamma, beta, out);
}